// tensor_80900003988126
// MI455X (gfx1250) — compile-verified
//
#include <hip/hip_runtime.h>
#include <cmath>

#define DH 129            // hat dimension (128 + 1)
#define NB 32             // batch
#define HD 64             // hidden
#define FUSED (DH*DH*DH)  // 2,146,689

typedef __attribute__((ext_vector_type(2))) float v2f;
typedef __attribute__((ext_vector_type(8))) float v8f;

__global__ void tfn_zero(float* __restrict__ y1) {
    int idx = blockIdx.x * blockDim.x + threadIdx.x;
    if (idx < NB * HD) y1[idx] = 0.0f;
}

__device__ __forceinline__ v8f wmma4(v2f a, v2f b, v8f c) {
    return __builtin_amdgcn_wmma_f32_16x16x4_f32(false, a, false, b,
                                                 (short)0, c, false, false);
}

__global__ __launch_bounds__(256) void tfn_main(
    const float* __restrict__ l1, const float* __restrict__ a1,
    const float* __restrict__ v1, const float* __restrict__ W1,
    float* __restrict__ y1_acc)
{
    __shared__ float sAh[NB * DH];
    __shared__ float sVh[NB * DH];
    __shared__ float sTh[NB * DH];
    __shared__ float sY1[NB * HD];

    const int tid = threadIdx.x;
    // Build hat matrices: column 0 = 1, columns 1..128 = input
    for (int idx = tid; idx < NB * DH; idx += 256) {
        int b = idx / DH, c = idx - b * DH;
        sAh[idx] = (c == 0) ? 1.0f : a1[b * 128 + c - 1];
        sVh[idx] = (c == 0) ? 1.0f : v1[b * 128 + c - 1];
        sTh[idx] = (c == 0) ? 1.0f : l1[b * 128 + c - 1];
    }
    for (int idx = tid; idx < NB * HD; idx += 256) sY1[idx] = 0.0f;
    __syncthreads();

    const int lane = tid & 31;
    const int wave = tid >> 5;          // wave32
    const int half = lane >> 4;         // 0: K sub 0..1 / M 0..7 ; 1: K 2..3 / M 8..15
    const int lm   = lane & 15;
    const int gw   = blockIdx.x * 8 + wave;
    const int nw   = gridDim.x * 8;

    v8f acc[2][4] = {};                 // [M-tile 16][N-tile 16] fp32 accumulators

    const float* thLo = &sTh[lm * DH];
    const float* thHi = &sTh[(lm + 16) * DH];

    for (int blk = gw; blk < DH * DH; blk += nw) {
        const int i = blk / DH, j = blk - i * DH;
        const float sLo = sAh[lm * DH + i]        * sVh[lm * DH + j];
        const float sHi = sAh[(lm + 16) * DH + i] * sVh[(lm + 16) * DH + j];
        const size_t base = (size_t)blk * DH;     // row of W1 for (i, j, k=0)

        // 32 full K=4 chunks covering k = 0..127
#pragma unroll 2
        for (int kc = 0; kc < 32; ++kc) {
            const int ka = kc * 4 + 2 * half;     // this lane's two K indices: ka, ka+1
            v2f aLo = { sLo * thLo[ka], sLo * thLo[ka + 1] };
            v2f aHi = { sHi * thHi[ka], sHi * thHi[ka + 1] };
            const float* p0 = W1 + (base + (size_t)ka) * HD + lm;  // K = ka
            const float* p1 = p0 + HD;                             // K = ka+1
#pragma unroll
            for (int t = 0; t < 4; ++t) {
                v2f bF = { p0[16 * t], p1[16 * t] };
                acc[0][t] = wmma4(aLo, bF, acc[0][t]);
                acc[1][t] = wmma4(aHi, bF, acc[1][t]);
            }
        }
        // Remainder chunk: only k = 128 is real; pad A with zeros, clamp B rows.
        {
            const int ka = 128 + 2 * half;                 // 128 (valid) or 130 (pad)
            const float t0 = (ka < DH) ? thLo[ka] : 0.0f;
            const float u0 = (ka < DH) ? thHi[ka] : 0.0f;
            v2f aLo = { sLo * t0, 0.0f };
            v2f aHi = { sHi * u0, 0.0f };
            size_t r0 = base + (size_t)ka; if (r0 > (size_t)(FUSED - 1)) r0 = FUSED - 1;
            size_t r1 = r0 + 1;            if (r1 > (size_t)(FUSED - 1)) r1 = FUSED - 1;
            const float* p0 = W1 + r0 * HD + lm;
            const float* p1 = W1 + r1 * HD + lm;
#pragma unroll
            for (int t = 0; t < 4; ++t) {
                v2f bF = { p0[16 * t], p1[16 * t] };
                acc[0][t] = wmma4(aLo, bF, acc[0][t]);
                acc[1][t] = wmma4(aHi, bF, acc[1][t]);
            }
        }
    }

    // Reduce this wave's 32x64 partial into LDS (ds_add_f32)
#pragma unroll
    for (int m = 0; m < 2; ++m)
#pragma unroll
        for (int t = 0; t < 4; ++t)
#pragma unroll
            for (int e = 0; e < 8; ++e) {
                int row = 16 * m + 8 * half + e;   // C/D layout: VGPR e -> M = e + 8*half
                int col = 16 * t + lm;
                atomicAdd(&sY1[row * HD + col], acc[m][t][e]);
            }
    __syncthreads();

    // One set of global fp32 atomics per workgroup
    for (int idx = tid; idx < NB * HD; idx += 256)
        unsafeAtomicAdd(&y1_acc[idx], sY1[idx]);
}

__global__ __launch_bounds__(256) void tfn_tail(
    const float* __restrict__ y1_acc, const float* __restrict__ b1,
    const float* __restrict__ W2, const float* __restrict__ b2,
    const float* __restrict__ W3, const float* __restrict__ b3,
    float* __restrict__ out)
{
    __shared__ float sy1[NB * HD];
    __shared__ float sy2[NB * HD];
    const int tid = threadIdx.x;
    for (int idx = tid; idx < NB * HD; idx += 256)
        sy1[idx] = y1_acc[idx] + b1[idx & 63];
    __syncthreads();
    for (int p = tid; p < NB * HD; p += 256) {
        int b = p >> 6, h = p & 63;
        float sum = b2[h];
#pragma unroll 4
        for (int k = 0; k < HD; ++k) sum += sy1[b * HD + k] * W2[k * HD + h];
        sy2[p] = tanhf(sum);
    }
    __syncthreads();
    if (tid < NB) {
        float sum = b3[0];
#pragma unroll 4
        for (int h = 0; h < HD; ++h) sum += sy2[tid * HD + h] * W3[h];
        out[tid] = fmaxf(sum, 0.0f);
    }
}

extern "C" void kernel_launch(void* const* d_in, const int* in_sizes, int n_in,
                              void* d_out, int out_size, void* d_ws, size_t ws_size,
                              hipStream_t stream) {
    (void)in_sizes; (void)n_in; (void)out_size; (void)ws_size;
    const float* l1 = (const float*)d_in[0];
    const float* a1 = (const float*)d_in[1];
    const float* v1 = (const float*)d_in[2];
    const float* W1 = (const float*)d_in[3];
    const float* b1 = (const float*)d_in[4];
    const float* W2 = (const float*)d_in[5];
    const float* b2 = (const float*)d_in[6];
    const float* W3 = (const float*)d_in[7];
    const float* b3 = (const float*)d_in[8];
    float* y1_acc = (float*)d_ws;   // 32*64 floats = 8 KB accumulator
    float* out = (float*)d_out;

    tfn_zero<<<8, 256, 0, stream>>>(y1_acc);
    // 256 workgroups x 8 waves = 2048 waves over 16641 (i,j) blocks (~8.1 each)
    tfn_main<<<256, 256, 0, stream>>>(l1, a1, v1, W1, y1_acc);
    tfn_tail<<<1, 256, 0, stream>>>(y1_acc, b1, W2, b2, W3, b3, out);
}